// YOLOv4loss_78889959293557
// MI455X (gfx1250) — compile-verified
//
#include <hip/hip_runtime.h>
#include <hip/hip_bf16.h>
#include <math.h>

#define TPB   256
#define NACC  7            // [0]=noobj_bce [1]=cnt_noobj [2]=obj_bce [3]=cnt_obj [4]=ciou [5]=box_sq [6]=cls_nll
#define YEPS  1e-7f

typedef __attribute__((ext_vector_type(2))) float v2f;
typedef __attribute__((ext_vector_type(8))) float v8f;

__device__ __forceinline__ float bce_logits(float l, float y) {
    // max(l,0) - l*y + log1p(exp(-|l|))
    return fmaxf(l, 0.f) - l * y + log1pf(expf(-fabsf(l)));
}

// ---------------- Stage 1: per-cell loss terms + block reduction ----------------
__global__ __launch_bounds__(TPB) void yolo_stage1(
    const float* __restrict__ pred,     // [ncells][85]
    const float* __restrict__ tgt,      // [ncells][6]
    const float* __restrict__ anchors,  // [3][2]
    float* __restrict__ partials,       // [NACC][nblocks]
    int ncells, int nblocks)
{
    __shared__ float smem[TPB * 9];     // stride 9 (coprime with 64 banks)
    const int tid  = threadIdx.x;
    const int cell = blockIdx.x * TPB + tid;
    const bool active = (cell < ncells);
    const int  c0 = active ? cell : 0;

    const float* p = pred + (size_t)c0 * 85u;
    const float* t = tgt  + (size_t)c0 * 6u;

    __builtin_prefetch(p, 0, 1);        // gfx1250: global_prefetch_b8

    const float flag    = t[0];
    const float m_obj   = (active && flag == 1.0f) ? 1.f : 0.f;
    const float m_noobj = (active && flag == 0.0f) ? 1.f : 0.f;
    const float p0 = p[0];

    float v0 = m_noobj * bce_logits(p0, flag);   // flag == 0 under the mask
    float v2 = 0.f, v4 = 0.f, v5 = 0.f, v6 = 0.f;

    if (m_obj != 0.f) {
        const int   ai = (c0 / (52 * 52)) % 3;
        const float aw = anchors[ai * 2 + 0];
        const float ah = anchors[ai * 2 + 1];
        const float t1 = t[1], t2 = t[2], t3 = t[3], t4 = t[4];
        const float p1 = p[1], p2 = p[2], p3 = p[3], p4 = p[4];

        const float bx = 1.f / (1.f + expf(-p1));
        const float by = 1.f / (1.f + expf(-p2));
        const float bw = expf(p3) * aw;
        const float bh = expf(p4) * ah;

        // boxes as corners
        const float b1x1 = bx - bw * 0.5f, b1y1 = by - bh * 0.5f;
        const float b1x2 = bx + bw * 0.5f, b1y2 = by + bh * 0.5f;
        const float b2x1 = t1 - t3 * 0.5f, b2y1 = t2 - t4 * 0.5f;
        const float b2x2 = t1 + t3 * 0.5f, b2y2 = t2 + t4 * 0.5f;

        // midpoint IoU (for objectness target)
        {
            const float iw = fmaxf(fminf(b1x2, b2x2) - fmaxf(b1x1, b2x1), 0.f);
            const float ih = fmaxf(fminf(b1y2, b2y2) - fmaxf(b1y1, b2y1), 0.f);
            const float inter = iw * ih;
            const float area1 = fabsf((b1x2 - b1x1) * (b1y2 - b1y1));
            const float area2 = fabsf((b2x2 - b2x1) * (b2y2 - b2y1));
            const float iou   = inter / (area1 + area2 - inter + 1e-6f);
            v2 = bce_logits(p0, iou);            // iou * flag, flag == 1 here
        }

        // CIoU loss
        {
            const float xx1 = fmaxf(b1x1, b2x1), yy1 = fmaxf(b1y1, b2y1);
            const float xx2 = fminf(b1x2, b2x2), yy2 = fminf(b1y2, b2y2);
            const float inter = fmaxf(xx2 - xx1, 0.f) * fmaxf(yy2 - yy1, 0.f);
            const float w1 = b1x2 - b1x1, h1 = b1y2 - b1y1;
            const float w2 = b2x2 - b2x1, h2 = b2y2 - b2y1;
            const float uni  = w1 * h1 + w2 * h2 - inter;
            const float iou2 = inter / (uni + YEPS);
            const float cw = fmaxf(b1x2, b2x2) - fminf(b1x1, b2x1);
            const float ch = fmaxf(b1y2, b2y2) - fminf(b1y1, b2y1);
            const float diag = cw * cw + ch * ch + YEPS;
            const float dx = (b1x1 + b1x2) * 0.5f - (b2x1 + b2x2) * 0.5f;
            const float dy = (b1y1 + b1y2) * 0.5f - (b2y1 + b2y2) * 0.5f;
            const float diou = 1.f - iou2 + (dx * dx + dy * dy) / diag;
            const float ar = atanf(w2 / (h2 + YEPS)) - atanf(w1 / (h1 + YEPS));
            const float vv = 0.405284734569351086f * ar * ar;   // 4/pi^2
            const float alpha = vv / (1.f - iou2 + vv + YEPS);
            v4 = diou + alpha * vv;
        }

        // coordinate (box) squared loss
        {
            const float lw = logf(1e-16f + t3 / aw);
            const float lh = logf(1e-16f + t4 / ah);
            const float d0 = bx - t1, d1 = by - t2, d2 = p3 - lw, d3 = p4 - lh;
            v5 = d0 * d0 + d1 * d1 + d2 * d2 + d3 * d3;
        }

        // class NLL via stable logsumexp (class logits only touched on obj cells)
        {
            const int label = (int)t[5];
            const float* pc = p + 5;
            float mx = pc[0];
            float plab = pc[0];
            for (int c = 1; c < 80; c++) {
                const float x = pc[c];
                mx = fmaxf(mx, x);
                if (c == label) plab = x;
            }
            float s = 0.f;
            for (int c = 0; c < 80; c++) s += expf(pc[c] - mx);
            v6 = mx + logf(s) - plab;
        }
    }

    float v[NACC] = { v0, m_noobj, v2, m_obj, v4, v5, v6 };
    #pragma unroll
    for (int j = 0; j < NACC; j++) smem[tid * 9 + j] = v[j];
    __syncthreads();
    for (int off = TPB / 2; off > 0; off >>= 1) {
        if (tid < off) {
            #pragma unroll
            for (int j = 0; j < NACC; j++)
                smem[tid * 9 + j] += smem[(tid + off) * 9 + j];
        }
        __syncthreads();
    }
    if (tid == 0) {
        #pragma unroll
        for (int j = 0; j < NACC; j++)
            partials[(size_t)j * nblocks + blockIdx.x] = smem[j];
    }
}

// ---------------- Stage 2: WMMA-chained deterministic fp32 reduction ----------------
// One wave (32 lanes). Each iteration feeds 32 partials into the A-matrix of
// V_WMMA_F32_16X16X4_F32 (lane L<16 -> A[L][0], lane L>=16 -> A[L-16][2], other
// K slots zero) and multiplies by an all-ones B, accumulating into C. Row sums
// replicate across D columns: lanes 0..15 hold rows 0..7, lanes 16..31 rows 8..15.
__global__ __launch_bounds__(32) void yolo_stage2(
    const float* __restrict__ partials,  // [NACC][nblocks]
    float* __restrict__ out,
    int nblocks)
{
    const int lane = threadIdx.x;
    __shared__ float red[2][NACC];
    float tloc[NACC];

#if __has_builtin(__builtin_amdgcn_wmma_f32_16x16x4_f32)
    v2f ones; ones.x = 1.f; ones.y = 1.f;
    v8f acc[NACC];
    #pragma unroll
    for (int j = 0; j < NACC; j++) acc[j] = (v8f)(0.f);

    for (int base = 0; base < nblocks; base += 32) {
        const int idx = base + lane;
        const float sc = (idx < nblocks) ? 1.f : 0.f;
        const int   ic = (idx < nblocks) ? idx : 0;     // clamp: no EXEC divergence
        #pragma unroll
        for (int j = 0; j < NACC; j++) {
            v2f a;
            a.x = partials[(size_t)j * nblocks + ic] * sc;
            a.y = 0.f;
            acc[j] = __builtin_amdgcn_wmma_f32_16x16x4_f32(
                false, a, false, ones, (short)0, acc[j], false, false);
        }
    }
    #pragma unroll
    for (int j = 0; j < NACC; j++) {
        float s = 0.f;
        #pragma unroll
        for (int r = 0; r < 8; r++) s += acc[j][r];
        tloc[j] = s;
    }
#else
#warning "wmma_f32_16x16x4_f32 builtin unavailable: scalar fallback in stage 2"
    #pragma unroll
    for (int j = 0; j < NACC; j++) tloc[j] = 0.f;
    for (int base = 0; base < nblocks; base += 32) {
        const int idx = base + lane;
        const float sc = (idx < nblocks) ? 1.f : 0.f;
        const int   ic = (idx < nblocks) ? idx : 0;
        #pragma unroll
        for (int j = 0; j < NACC; j++)
            tloc[j] += partials[(size_t)j * nblocks + ic] * sc;
    }
    // fold lanes 1..15 into lane 0 and 17..31 into lane 16 via LDS
    __shared__ float fold[32 * NACC];
    #pragma unroll
    for (int j = 0; j < NACC; j++) fold[lane * NACC + j] = tloc[j];
    __syncthreads();
    if (lane == 0 || lane == 16) {
        #pragma unroll
        for (int j = 0; j < NACC; j++) {
            float s = 0.f;
            for (int l = lane; l < lane + 16; l++) s += fold[l * NACC + j];
            tloc[j] = s;
        }
    }
#endif

    if (lane == 0)  { for (int j = 0; j < NACC; j++) red[0][j] = tloc[j]; }
    if (lane == 16) { for (int j = 0; j < NACC; j++) red[1][j] = tloc[j]; }
    __syncthreads();

    if (lane == 0) {
        float S[NACC];
        #pragma unroll
        for (int j = 0; j < NACC; j++) S[j] = red[0][j] + red[1][j];
        const float cno = fmaxf(S[1], 1.f);
        const float cob = fmaxf(S[3], 1.f);
        const float no_object_loss = S[0] / cno;
        const float object_loss    = S[2] / cob;
        const float ciou_loss      = S[4] / cob;
        const float box_loss       = (S[5] / cob) * 0.25f;
        const float class_loss     = S[6] / cob;
        out[0] = 10.f * box_loss + 10.f * object_loss +
                 1.f * no_object_loss + 1.f * class_loss + ciou_loss;
    }
}

extern "C" void kernel_launch(void* const* d_in, const int* in_sizes, int n_in,
                              void* d_out, int out_size, void* d_ws, size_t ws_size,
                              hipStream_t stream) {
    const float* pred    = (const float*)d_in[0];
    const float* tgt     = (const float*)d_in[1];
    const float* anchors = (const float*)d_in[2];
    float* out = (float*)d_out;

    const int ncells  = in_sizes[0] / 85;            // B*A*S*S
    const int nblocks = (ncells + TPB - 1) / TPB;
    float* partials = (float*)d_ws;                  // NACC * nblocks floats

    yolo_stage1<<<nblocks, TPB, 0, stream>>>(pred, tgt, anchors, partials, ncells, nblocks);
    yolo_stage2<<<1, 32, 0, stream>>>(partials, out, nblocks);
}